// ProcessorCNN_197568495945
// MI455X (gfx1250) — compile-verified
//
#include <hip/hip_runtime.h>

#define NPTS 16384
#define DIM  512
#define KNBR 6
#define CHUNK 2048

typedef float v2f __attribute__((ext_vector_type(2)));
typedef float v4f __attribute__((ext_vector_type(4)));
typedef float v8f __attribute__((ext_vector_type(8)));

// ---------------------------------------------------------------------------
// Fibonacci-sphere positions (f32, matching jax op order)
// ---------------------------------------------------------------------------
__global__ void pos_kernel(float* __restrict__ px, float* __restrict__ py,
                           float* __restrict__ pz) {
  int i = blockIdx.x * blockDim.x + threadIdx.x;
  if (i >= NPTS) return;
  const float golden = (1.0f + sqrtf(5.0f)) * 0.5f;
  float theta = (6.28318530717958647692f * (float)i) / golden;
  float t     = (2.0f * ((float)i + 0.5f)) / (float)NPTS;
  float phi   = acosf(1.0f - t);
  float sp = sinf(phi);
  px[i] = cosf(theta) * sp;
  py[i] = sinf(theta) * sp;
  pz[i] = cosf(phi);
}

// ---------------------------------------------------------------------------
// Brute-force geodesic 6-NN. arccos is monotone decreasing, so nearest
// geodesic == largest clipped dot; ties -> lower index (lax.top_k order).
// Keep top-7 (self has dot ~1.0 -> rank 0), emit entries 1..6.
// ---------------------------------------------------------------------------
__global__ void knn_kernel(const float* __restrict__ px,
                           const float* __restrict__ py,
                           const float* __restrict__ pz,
                           int* __restrict__ nbr) {
  __shared__ float sx[CHUNK], sy[CHUNK], sz[CHUNK];
  int i = blockIdx.x * blockDim.x + threadIdx.x;
  float mx = px[i], my = py[i], mz = pz[i];
  float bd[7]; int bi[7];
#pragma unroll
  for (int k = 0; k < 7; ++k) { bd[k] = -2.0f; bi[k] = 0x7fffffff; }

  for (int base = 0; base < NPTS; base += CHUNK) {
    __syncthreads();
    for (int t = threadIdx.x; t < CHUNK; t += blockDim.x) {
      sx[t] = px[base + t]; sy[t] = py[base + t]; sz[t] = pz[base + t];
    }
    __syncthreads();
    for (int j = 0; j < CHUNK; ++j) {
      float d = mx * sx[j];
      d = fmaf(my, sy[j], d);
      d = fmaf(mz, sz[j], d);
      d = fminf(fmaxf(d, -1.0f), 1.0f);
      int cj = base + j;
      if ((d > bd[6]) || (d == bd[6] && cj < bi[6])) {
        bd[6] = d; bi[6] = cj;
#pragma unroll
        for (int k = 6; k > 0; --k) {
          bool sw = (bd[k] > bd[k - 1]) || (bd[k] == bd[k - 1] && bi[k] < bi[k - 1]);
          if (sw) {
            float td = bd[k]; bd[k] = bd[k - 1]; bd[k - 1] = td;
            int   ti = bi[k]; bi[k] = bi[k - 1]; bi[k - 1] = ti;
          }
        }
      }
    }
  }
#pragma unroll
  for (int k = 0; k < KNBR; ++k) nbr[i * KNBR + k] = bi[k + 1];
}

// ---------------------------------------------------------------------------
// neigh_mean: nm[row] = mean over 6 neighbor rows of h. float4 per thread.
// ---------------------------------------------------------------------------
__global__ void neigh_mean_kernel(const float* __restrict__ h,
                                  const int* __restrict__ nbr,
                                  float* __restrict__ nm) {
  int row = blockIdx.x;
  int t   = threadIdx.x;  // 0..127, one float4 each
  const int* nb = nbr + row * KNBR;
  v4f s = {};
#pragma unroll
  for (int k = 0; k < KNBR; ++k) {
    const v4f* p = (const v4f*)(h + (size_t)nb[k] * DIM) + t;
    s += *p;
  }
  s = s / 6.0f;
  ((v4f*)(nm + (size_t)row * DIM))[t] = s;
}

// ---------------------------------------------------------------------------
// Fused dual-GEMM + residual + bias + LayerNorm + ReLU, f32 WMMA.
// Block = 256 threads (8 waves), tile = 32 rows x 512 cols.
// Wave w owns cols [64w, 64w+64) = 4 col-tiles, and BOTH 16-row tiles:
// each B fragment feeds 2 WMMAs, each A fragment feeds 4 WMMAs
// (10 loads -> 8 WMMAs per k-step).
//
// V_WMMA_F32_16X16X4_F32 fragment layouts (ISA 7.12.2):
//   A 16x4:  lane l -> m = l&15, holds K = kbase+2*(l>>4) (+1 in vgpr1)
//   B 4x16:  lane l -> n = l&15, holds K = kbase+2*(l>>4) (+1 in vgpr1)
//   C/D:     vgpr r, lane l -> row = r + 8*(l>>4), col = l&15
// ---------------------------------------------------------------------------
__global__ __launch_bounds__(256)
void gnn_step_kernel(const float* __restrict__ h_in,
                     const float* __restrict__ nm,
                     const float* __restrict__ Ws, const float* __restrict__ bs,
                     const float* __restrict__ Wn, const float* __restrict__ bn,
                     const float* __restrict__ lnS, const float* __restrict__ lnO,
                     float* __restrict__ h_out) {
  __shared__ float sS[32][8], sQ[32][8], sMu[32], sRs[32];
  const int lane    = threadIdx.x & 31;
  const int wave    = threadIdx.x >> 5;   // 0..7
  const int rowbase = blockIdx.x * 32;
  const int colbase = wave * 64;
  const int half    = lane >> 4;
  const int n16     = lane & 15;

  v8f acc[2][4];
  v8f vzero = {};
#pragma unroll
  for (int rt = 0; rt < 2; ++rt)
#pragma unroll
    for (int t = 0; t < 4; ++t) acc[rt][t] = vzero;

#pragma unroll
  for (int phase = 0; phase < 2; ++phase) {
    const float* A = phase ? nm : h_in;
    const float* W = phase ? Wn : Ws;
    const float* arow0 = A + (size_t)(rowbase + n16) * DIM + 2 * half;
    const float* arow1 = arow0 + (size_t)16 * DIM;
    const float* wcol  = W + (size_t)(2 * half) * DIM + colbase + n16;
    for (int k = 0; k < DIM; k += 4) {
      v2f a0 = *(const v2f*)(arow0 + k);
      v2f a1 = *(const v2f*)(arow1 + k);
      const float* wk = wcol + (size_t)k * DIM;
#pragma unroll
      for (int t = 0; t < 4; ++t) {
        v2f b;
        b.x = wk[t * 16];
        b.y = wk[t * 16 + DIM];
        acc[0][t] = __builtin_amdgcn_wmma_f32_16x16x4_f32(
            false, a0, false, b, (short)0, acc[0][t], false, false);
        acc[1][t] = __builtin_amdgcn_wmma_f32_16x16x4_f32(
            false, a1, false, b, (short)0, acc[1][t], false, false);
      }
    }
  }

  // x = h + selfT + neighT + bs + bn
#pragma unroll
  for (int rt = 0; rt < 2; ++rt)
#pragma unroll
    for (int t = 0; t < 4; ++t) {
      int col = colbase + t * 16 + n16;
      float bsum = bs[col] + bn[col];
#pragma unroll
      for (int r = 0; r < 8; ++r) {
        int m = rt * 16 + r + half * 8;
        acc[rt][t][r] =
            acc[rt][t][r] + h_in[(size_t)(rowbase + m) * DIM + col] + bsum;
      }
    }

  // per-row sum / sumsq over this wave's 64 cols
  float ps[2][8], pq[2][8];
#pragma unroll
  for (int rt = 0; rt < 2; ++rt)
#pragma unroll
    for (int r = 0; r < 8; ++r) { ps[rt][r] = 0.f; pq[rt][r] = 0.f; }
#pragma unroll
  for (int rt = 0; rt < 2; ++rt)
#pragma unroll
    for (int t = 0; t < 4; ++t)
#pragma unroll
      for (int r = 0; r < 8; ++r) {
        float x = acc[rt][t][r];
        ps[rt][r] += x;
        pq[rt][r] += x * x;
      }

#pragma unroll
  for (int mask = 1; mask < 16; mask <<= 1) {
#pragma unroll
    for (int rt = 0; rt < 2; ++rt)
#pragma unroll
      for (int r = 0; r < 8; ++r) {
        ps[rt][r] += __shfl_xor(ps[rt][r], mask, 32);
        pq[rt][r] += __shfl_xor(pq[rt][r], mask, 32);
      }
  }
  if (n16 == 0) {
#pragma unroll
    for (int rt = 0; rt < 2; ++rt)
#pragma unroll
      for (int r = 0; r < 8; ++r) {
        int m = rt * 16 + r + half * 8;
        sS[m][wave] = ps[rt][r];
        sQ[m][wave] = pq[rt][r];
      }
  }
  __syncthreads();
  if (threadIdx.x < 32) {
    int m = threadIdx.x;
    float S = 0.f, Q = 0.f;
#pragma unroll
    for (int w = 0; w < 8; ++w) { S += sS[m][w]; Q += sQ[m][w]; }
    float mu  = S * (1.0f / 512.0f);
    float var = fmaxf(Q * (1.0f / 512.0f) - mu * mu, 0.0f);
    sMu[m] = mu;
    sRs[m] = rsqrtf(var + 1e-5f);
  }
  __syncthreads();

#pragma unroll
  for (int rt = 0; rt < 2; ++rt)
#pragma unroll
    for (int t = 0; t < 4; ++t) {
      int col = colbase + t * 16 + n16;
      float sc = lnS[col], of = lnO[col];
#pragma unroll
      for (int r = 0; r < 8; ++r) {
        int m = rt * 16 + r + half * 8;
        float y = sc * (acc[rt][t][r] - sMu[m]) * sRs[m] + of;
        h_out[(size_t)(rowbase + m) * DIM + col] = fmaxf(y, 0.0f);
      }
    }
}

// ---------------------------------------------------------------------------
extern "C" void kernel_launch(void* const* d_in, const int* in_sizes, int n_in,
                              void* d_out, int out_size, void* d_ws, size_t ws_size,
                              hipStream_t stream) {
  const float* h0   = (const float*)d_in[0];
  const float* ws_w = (const float*)d_in[1];
  const float* ws_b = (const float*)d_in[2];
  const float* wn_w = (const float*)d_in[3];
  const float* wn_b = (const float*)d_in[4];
  const float* ln_s = (const float*)d_in[5];
  const float* ln_o = (const float*)d_in[6];
  float* out = (float*)d_out;

  char* ws = (char*)d_ws;
  int*   nbr  = (int*)(ws);                                // 393216 B
  float* px   = (float*)(ws + 393216);                     // 64 KB
  float* py   = (float*)(ws + 393216 + 65536);
  float* pz   = (float*)(ws + 393216 + 131072);
  float* nm   = (float*)(ws + 589824);                     // 32 MB
  float* htmp = (float*)(ws + 589824 + 33554432);          // 32 MB

  pos_kernel<<<NPTS / 256, 256, 0, stream>>>(px, py, pz);
  knn_kernel<<<NPTS / 256, 256, 0, stream>>>(px, py, pz, nbr);

  const float* hc = h0;
  float* bufs[3] = {out, htmp, out};
  for (int i = 0; i < 3; ++i) {
    neigh_mean_kernel<<<NPTS, 128, 0, stream>>>(hc, nbr, nm);
    gnn_step_kernel<<<NPTS / 32, 256, 0, stream>>>(
        hc, nm,
        ws_w + (size_t)i * DIM * DIM, ws_b + (size_t)i * DIM,
        wn_w + (size_t)i * DIM * DIM, wn_b + (size_t)i * DIM,
        ln_s + (size_t)i * DIM, ln_o + (size_t)i * DIM, bufs[i]);
    hc = bufs[i];
  }
}